// MessagePassingBlock_8864812499249
// MI455X (gfx1250) — compile-verified
//
#include <hip/hip_runtime.h>
#include <hip/hip_bf16.h>

#define HID 384

typedef __attribute__((ext_vector_type(2))) float v2f;
typedef __attribute__((ext_vector_type(8))) float v8f;

// ---------------------------------------------------------------------------
// Degree accumulation: deg[row[e]] += 1
// ---------------------------------------------------------------------------
__global__ __launch_bounds__(256) void gcn_degree(const int* __restrict__ row,
                                                  float* __restrict__ deg,
                                                  int nEdges) {
    int e = blockIdx.x * blockDim.x + threadIdx.x;
    if (e < nEdges) {
        unsafeAtomicAdd(&deg[row[e]], 1.0f);
    }
}

// dis[i] = 1/sqrt(deg[i])   (deg >= 1 by construction)
__global__ __launch_bounds__(256) void gcn_rsqrt(float* __restrict__ dis, int n) {
    int i = blockIdx.x * blockDim.x + threadIdx.x;
    if (i < n) {
        dis[i] = 1.0f / sqrtf(dis[i]);
    }
}

// norm[e] = dis[row[e]] * dis[col[e]]
__global__ __launch_bounds__(256) void gcn_norm(const int* __restrict__ row,
                                                const int* __restrict__ col,
                                                const float* __restrict__ dis,
                                                float* __restrict__ norm,
                                                int nEdges) {
    int e = blockIdx.x * blockDim.x + threadIdx.x;
    if (e < nEdges) {
        norm[e] = dis[row[e]] * dis[col[e]];
    }
}

// ---------------------------------------------------------------------------
// GEMM: Y[i][j] = bias[j] + sum_k X[i][k] * W[j][k]     (y = x @ W^T + b)
// One wave computes a 16-row x 64-col tile via 4 accumulators of
// V_WMMA_F32_16X16X4_F32, stepping K by 4.
//
// A fragment (16x4 f32): lanes 0-15 rows M=0..15 hold K=0,1 in the two VGPRs;
//                        lanes 16-31 hold K=2,3.
// B fragment (4x16 f32): VGPR0 = row K=0 (lanes 0-15, N=lane) / K=2 (lanes
//                        16-31); VGPR1 = K=1 / K=3.  B[k][n] = W[n][k].
// C/D (16x16 f32, 8 VGPRs): VGPR v, lane L -> (M = v + 8*(L>>4), N = L&15).
// ---------------------------------------------------------------------------
__global__ __launch_bounds__(256) void gcn_gemm(const float* __restrict__ X,
                                                const float* __restrict__ W,
                                                const float* __restrict__ bias,
                                                float* __restrict__ Y,
                                                int nRows) {
    const int lane    = threadIdx.x & 31;
    const int waveInB = threadIdx.x >> 5;
    const int wavesPB = blockDim.x >> 5;
    const int waveId  = blockIdx.x * wavesPB + waveInB;

    const int colStrips = HID / 64;            // 6
    const int rowTile   = waveId / colStrips;
    const int n0        = (waveId % colStrips) * 64;
    if (rowTile * 16 >= nRows) return;         // wave-uniform: EXEC stays all-1s
    const int m0 = rowTile * 16;

    const int lhalf = lane >> 4;               // 0/1 -> K subpair
    const int l15   = lane & 15;               // row (A) / col (B,C,D)

    // Accumulators initialized with bias: value at (M,N) depends only on N.
    const float bv0 = bias[n0 +  0 + l15];
    const float bv1 = bias[n0 + 16 + l15];
    const float bv2 = bias[n0 + 32 + l15];
    const float bv3 = bias[n0 + 48 + l15];
    v8f acc0, acc1, acc2, acc3;
#pragma unroll
    for (int v = 0; v < 8; ++v) {
        acc0[v] = bv0; acc1[v] = bv1; acc2[v] = bv2; acc3[v] = bv3;
    }

    const float* xrow = X + (size_t)(m0 + l15) * HID + lhalf * 2;
    const float* w0   = W + (size_t)(n0 +  0 + l15) * HID + lhalf * 2;
    const float* w1   = W + (size_t)(n0 + 16 + l15) * HID + lhalf * 2;
    const float* w2   = W + (size_t)(n0 + 32 + l15) * HID + lhalf * 2;
    const float* w3   = W + (size_t)(n0 + 48 + l15) * HID + lhalf * 2;

#pragma unroll 4
    for (int k = 0; k < HID; k += 4) {
        v2f a  = *(const v2f*)(xrow + k);
        v2f bb0 = *(const v2f*)(w0 + k);
        v2f bb1 = *(const v2f*)(w1 + k);
        v2f bb2 = *(const v2f*)(w2 + k);
        v2f bb3 = *(const v2f*)(w3 + k);
        acc0 = __builtin_amdgcn_wmma_f32_16x16x4_f32(false, a, false, bb0,
                                                     (short)0, acc0, false, false);
        acc1 = __builtin_amdgcn_wmma_f32_16x16x4_f32(false, a, false, bb1,
                                                     (short)0, acc1, false, false);
        acc2 = __builtin_amdgcn_wmma_f32_16x16x4_f32(false, a, false, bb2,
                                                     (short)0, acc2, false, false);
        acc3 = __builtin_amdgcn_wmma_f32_16x16x4_f32(false, a, false, bb3,
                                                     (short)0, acc3, false, false);
    }

    // Store: VGPR v of each acc holds row M = v + 8*lhalf, col = n0 + sub*16 + l15.
    float* yb = Y + (size_t)m0 * HID + n0;
#pragma unroll
    for (int v = 0; v < 8; ++v) {
        const int M = v + 8 * lhalf;
        yb[(size_t)M * HID +  0 + l15] = acc0[v];
        yb[(size_t)M * HID + 16 + l15] = acc1[v];
        yb[(size_t)M * HID + 32 + l15] = acc2[v];
        yb[(size_t)M * HID + 48 + l15] = acc3[v];
    }
}

// ---------------------------------------------------------------------------
// Aggregate: Out[col[e]] += norm[e] * Y[row[e]]   (one block per edge)
// ---------------------------------------------------------------------------
__global__ __launch_bounds__(HID) void gcn_aggregate(const float* __restrict__ Y,
                                                     const int* __restrict__ row,
                                                     const int* __restrict__ col,
                                                     const float* __restrict__ norm,
                                                     float* __restrict__ Out,
                                                     int nEdges) {
    const int e = blockIdx.x;
    if (e >= nEdges) return;
    const int r   = row[e];
    const int c   = col[e];
    const float s = norm[e];
    const int t   = threadIdx.x;
    const float v = s * Y[(size_t)r * HID + t];
    unsafeAtomicAdd(&Out[(size_t)c * HID + t], v);
}

// ---------------------------------------------------------------------------
extern "C" void kernel_launch(void* const* d_in, const int* in_sizes, int n_in,
                              void* d_out, int out_size, void* d_ws, size_t ws_size,
                              hipStream_t stream) {
    const float* x  = (const float*)d_in[0];
    const int*   ei = (const int*)d_in[1];     // [2, E] flat, int32
    const float* W1 = (const float*)d_in[2];
    const float* b1 = (const float*)d_in[3];
    const float* W2 = (const float*)d_in[4];
    const float* b2 = (const float*)d_in[5];

    const int nNodes = in_sizes[0] / HID;      // 100000
    const int nEdges = in_sizes[1] / 2;        // 200000
    const int* row = ei;
    const int* col = ei + nEdges;

    // Workspace layout (floats): dis[N] | norm[E] | bufA[N*H] | bufB[N*H]
    float* ws   = (float*)d_ws;
    float* dis  = ws;
    float* nrm  = dis + nNodes;
    float* bufA = nrm + nEdges;
    float* bufB = bufA + (size_t)nNodes * HID;

    const size_t featBytes = (size_t)nNodes * HID * sizeof(float);

    // --- edge normalization (shared by both layers) ---
    hipMemsetAsync(dis, 0, (size_t)nNodes * sizeof(float), stream);
    gcn_degree<<<(nEdges + 255) / 256, 256, 0, stream>>>(row, dis, nEdges);
    gcn_rsqrt<<<(nNodes + 255) / 256, 256, 0, stream>>>(dis, nNodes);
    gcn_norm<<<(nEdges + 255) / 256, 256, 0, stream>>>(row, col, dis, nrm, nEdges);

    const int waveTiles = (nNodes / 16) * (HID / 64);
    const int gemmBlocks = (waveTiles + 7) / 8;   // 8 waves (256 thr) per block

    // --- layer 1 ---
    gcn_gemm<<<gemmBlocks, 256, 0, stream>>>(x, W1, b1, bufA, nNodes);
    hipMemsetAsync(bufB, 0, featBytes, stream);
    gcn_aggregate<<<nEdges, HID, 0, stream>>>(bufA, row, col, nrm, bufB, nEdges);

    // --- layer 2 ---
    gcn_gemm<<<gemmBlocks, 256, 0, stream>>>(bufB, W2, b2, bufA, nNodes);
    hipMemsetAsync(d_out, 0, featBytes, stream);
    gcn_aggregate<<<nEdges, HID, 0, stream>>>(bufA, row, col, nrm,
                                              (float*)d_out, nEdges);
}